// EdgeDecoder_45054206935079
// MI455X (gfx1250) — compile-verified
//
#include <hip/hip_runtime.h>
#include <hip/hip_bf16.h>

// EdgeDecoder: out[i] = h[src[i]]·w_src + h[dst[i]]·w_dst + e[i]·w_e + bias
// Two-phase, bandwidth-optimal:
//   Phase 1: q[n] = h[n]·w_src, r[n] = h[n]·w_dst   (25.6 MB read, 400 KB written)
//   Phase 2: out[i] = q[src[i]] + r[dst[i]] + e[i]·w_e + bias  (327 MB stream)
// Dot products on V_WMMA_F32_16X16X4_F32 with WEIGHT in A, DATA in B:
//   - A row m only affects C row m, so unused A rows need no masking.
//   - C row 0 (acc[0], lanes 0-15) = one dot product per lane -> parallel epilogue.

constexpr int kD     = 128;
constexpr int kNodes = 50000;
constexpr int kEdges = 640000;

typedef __attribute__((ext_vector_type(2))) float v2f;
typedef __attribute__((ext_vector_type(8))) float v8f;

// ---------------------------------------------------------------------------
// Phase 1: 16 nodes per wave.
//   A: row 0 = w_src chunk, row 1 = w_dst chunk (lane c15 supplies row c15;
//      lanes >= 2 redundantly load w_src -- harmless, rows 2-15 of C ignored).
//   B: column n = h[nodeBase+n] K-chunk.
//   C: acc[0] lane n = q, acc[1] lane n = r.
// A-lane layout (f32 16x4): lane L<16 -> row L K-pair (k,k+1); lane L>=16 ->
// row L-16 K-pair (k+2,k+3). B mirrors it with N across lanes.
// ---------------------------------------------------------------------------
__global__ __launch_bounds__(256) void node_proj_wmma(
    const float* __restrict__ h, const float* __restrict__ weight,
    float* __restrict__ q, float* __restrict__ r)
{
  const int lane = threadIdx.x & 31;
  const int wave = blockIdx.x * (blockDim.x >> 5) + (threadIdx.x >> 5);
  const int tiles = kNodes / 16;            // 3125 exactly
  if (wave >= tiles) return;                // wave-uniform: EXEC stays all-1s
  const int nodeBase = wave * 16;

  const int c15  = lane & 15;
  const int koff = (lane >> 4) * 2;         // K sub-offset for this half-wave

  // A operand: this lane's matrix row (row 0 = w_src, row 1 = w_dst).
  const float* aptr = ((c15 == 1) ? (weight + kD) : weight) + koff;
  // B operand: this lane's data column (node nodeBase+c15).
  const float* bptr = h + (size_t)(nodeBase + c15) * kD + koff;

  v8f acc = {};
  #pragma unroll 8
  for (int k = 0; k < kD; k += 4) {
    v2f a = *(const v2f*)(aptr + k);        // broadcast-ish, L0-resident
    v2f b = *(const v2f*)(bptr + k);
    acc = __builtin_amdgcn_wmma_f32_16x16x4_f32(
        /*neg_a=*/false, a, /*neg_b=*/false, b,
        /*c_mod=*/(short)0, acc, /*reuse_a=*/false, /*reuse_b=*/false);
  }

  if (lane < 16) {                          // one node per lane
    q[nodeBase + lane] = acc[0];
    r[nodeBase + lane] = acc[1];
  }
}

// ---------------------------------------------------------------------------
// Phase 2: 16 edges per wave.
//   A: every lane loads the w_e chunk (rows of C all equal; we read row 0).
//   B: column n = e[edgeBase+n] K-chunk.
//   C: acc[0] lane n = e[edge n]·w_e.
// Epilogue: lanes 0-15, one edge each: + q[src] + r[dst] + bias, store.
// ---------------------------------------------------------------------------
__global__ __launch_bounds__(256) void edge_decode_wmma(
    const float* __restrict__ e, const int* __restrict__ src,
    const int* __restrict__ dst, const float* __restrict__ weight,
    const float* __restrict__ bias, const float* __restrict__ q,
    const float* __restrict__ r, float* __restrict__ out)
{
  const int lane = threadIdx.x & 31;
  const int wave = blockIdx.x * (blockDim.x >> 5) + (threadIdx.x >> 5);
  const int edgeBase = wave * 16;           // grid sized exactly: 40000 tiles

  const int c15  = lane & 15;
  const int koff = (lane >> 4) * 2;

  const float* aptr = weight + 2 * kD + koff;                 // w_e
  const float* bptr = e + (size_t)(edgeBase + c15) * kD + koff;

  v8f acc = {};
  #pragma unroll 8
  for (int k = 0; k < kD; k += 4) {
    v2f a = *(const v2f*)(aptr + k);
    v2f b = *(const v2f*)(bptr + k);
    acc = __builtin_amdgcn_wmma_f32_16x16x4_f32(
        false, a, false, b, (short)0, acc, false, false);
  }

  if (lane < 16) {                          // one edge per lane
    const int ei = edgeBase + lane;
    out[ei] = acc[0] + q[src[ei]] + r[dst[ei]] + bias[0];
  }
}

extern "C" void kernel_launch(void* const* d_in, const int* in_sizes, int n_in,
                              void* d_out, int out_size, void* d_ws, size_t ws_size,
                              hipStream_t stream) {
  const float* h      = (const float*)d_in[0];
  const float* e      = (const float*)d_in[1];
  const int*   src    = (const int*)d_in[2];
  const int*   dst    = (const int*)d_in[3];
  const float* weight = (const float*)d_in[4];
  const float* bias   = (const float*)d_in[5];
  float* out = (float*)d_out;

  float* q = (float*)d_ws;                  // kNodes floats
  float* r = q + kNodes;                    // kNodes floats (400 KB total)

  // Phase 1: 3125 node tiles, 8 waves/block -> 391 blocks.
  node_proj_wmma<<<(kNodes / 16 + 7) / 8, 256, 0, stream>>>(h, weight, q, r);
  // Phase 2: 40000 edge tiles, 8 waves/block -> 5000 blocks.
  edge_decode_wmma<<<(kEdges / 16) / 8, 256, 0, stream>>>(
      e, src, dst, weight, bias, q, r, out);
}